// GAT_77902116815247
// MI455X (gfx1250) — compile-verified
//
#include <hip/hip_runtime.h>
#include <hip/hip_bf16.h>

// Problem constants (match reference)
#define NNODES 10000
#define NEDGES 320000
#define FIN    128
#define CCH    128           // per-head channels
#define HEADS  4
#define HC     512           // HEADS*CCH
#define LAYERS 3
#define GRAPHS 64
#define DNN    512
#define NCLS   10

#define LDS_PAD 4            // floats of row padding -> lane bank stride 4
#define MAXK    512

typedef __attribute__((ext_vector_type(2))) float v2f;
typedef __attribute__((ext_vector_type(8))) float v8f;

// ---------- monotone float <-> u32 encoding (for atomic max on floats) ----------
__device__ __forceinline__ unsigned enc_f32(float f) {
    unsigned u = __float_as_uint(f);
    return (u & 0x80000000u) ? ~u : (u | 0x80000000u);
}
__device__ __forceinline__ float dec_f32(unsigned u) {
    if (u == 0u) return -__builtin_inff();          // never-written sentinel
    return (u & 0x80000000u) ? __uint_as_float(u & 0x7FFFFFFFu)
                             : __uint_as_float(~u);
}

// hardware fp32 atomic add (global_atomic_add_f32, no CAS loop)
__device__ __forceinline__ void atomic_fadd(float* p, float v) {
    unsafeAtomicAdd(p, v);
}

// ---------- CDNA5 async copy: 16B global -> LDS, tracked by ASYNCcnt ----------
__device__ __forceinline__ void async_copy16(unsigned lds_addr, const void* gaddr) {
    asm volatile("global_load_async_to_lds_b128 %0, %1, off"
                 :: "v"(lds_addr), "v"(gaddr) : "memory");
}
__device__ __forceinline__ void wait_async0() {
    asm volatile("s_wait_asynccnt 0x0" ::: "memory");
}

// ---------- generic zero-fill ----------
__global__ void zero_u32(unsigned* __restrict__ p, int n) {
    int i = blockIdx.x * blockDim.x + threadIdx.x;
    if (i < n) p[i] = 0u;
}

// ---------- fp32 WMMA GEMM: C[M,Nc] = act(A[M,K] @ B[K,Nc] + bias) + resid ----------
// One block owns a 32-row panel (two 16-row strips) and ALL Nc columns.
// blockDim = (32, Nc/64): each wave32 computes 2x4 = 8 independent 16x16
// accumulator tiles, so each 8-load B fragment set feeds 8 V_WMMA_F32_16X16X4_F32.
// Both A strips are staged in LDS once per block via GLOBAL_LOAD_ASYNC_TO_LDS_B128
// (ASYNCcnt) with a 4-float row pad for conflict-free ds_load_b64 fragment reads.
// B fragments are software-pipelined one k-step ahead (pointer-bump prefetch,
// last k-step peeled) to overlap global loads with the WMMA chain.
// M % 16 == 0, K % 4 == 0 and K >= 8, Nc % 64 == 0 at all call sites;
// a trailing 16-row block (M % 32 == 16) disables its second strip's stores.
__global__ void wmma_gemm_f32(const float* __restrict__ A, const float* __restrict__ B,
                              const float* __restrict__ bias, const float* __restrict__ resid,
                              float* __restrict__ C, int M, int K, int Nc, int do_relu)
{
    __shared__ float As[32 * (MAXK + LDS_PAD)];

    const int row0 = blockIdx.x * 32;
    const bool two = (row0 + 16) < M;               // second strip valid?
    const int nrows = two ? 32 : 16;
    const int lane = threadIdx.x;                   // 0..31 (wave32)
    const int half = lane >> 4;                     // 0: lanes 0-15, 1: lanes 16-31
    const int lm   = lane & 15;
    const int tid  = threadIdx.y * 32 + lane;
    const int T    = blockDim.y * 32;
    const int ldk  = K + LDS_PAD;

    __builtin_prefetch(B, 0, 1);                    // global_prefetch_b8 (L2-hot weights)

    // ---- stage A panel (nrows*K floats, contiguous) into padded LDS rows ----
    {
        const char* gbase = (const char*)(A + (size_t)row0 * K);
        unsigned lds_base = (unsigned)(size_t)(void*)As;   // low 32 bits = LDS offset
        int chunks = nrows * K / 4;                         // 16B chunks in panel
        for (int i = tid; i < chunks; i += T) {
            int f0  = i << 2;                               // float index in panel
            int row = f0 / K;
            int col = f0 - row * K;
            async_copy16(lds_base + (unsigned)(row * ldk + col) * 4u,
                         gbase + (size_t)i * 16);
        }
        wait_async0();
        __syncthreads();
    }

    // ---- 8 tiles per wave: 2 row strips x 4 column tiles ----
    const int colw = threadIdx.y * 64;
    const float* afp0 = &As[lm * ldk + 2 * half];
    const float* afp1 = &As[(16 + lm) * ldk + 2 * half];

    auto loadB = [&](const float* p, v2f bb[4]) {
#pragma unroll
        for (int t = 0; t < 4; ++t) {
            bb[t].x = p[t * 16];
            bb[t].y = p[t * 16 + Nc];
        }
    };

    v8f acc[8] = {};
    const float* bp = B + (size_t)(2 * half) * Nc + colw + lm;
    v2f bcur[4];
    loadB(bp, bcur);
    bp += (size_t)4 * Nc;                            // next k-step base

    // steady state: prefetch k0+4 with a pure pointer bump (always in bounds)
    for (int k0 = 0; k0 < K - 4; k0 += 4) {
        v2f bnx[4];
        loadB(bp, bnx);
        v2f a0 = *(const v2f*)(afp0 + k0);           // ds_load_b64, conflict-free
        v2f a1 = *(const v2f*)(afp1 + k0);
#pragma unroll
        for (int t = 0; t < 4; ++t) {
            acc[t]     = __builtin_amdgcn_wmma_f32_16x16x4_f32(
                             false, a0, false, bcur[t], (short)0, acc[t],     false, false);
            acc[4 + t] = __builtin_amdgcn_wmma_f32_16x16x4_f32(
                             false, a1, false, bcur[t], (short)0, acc[4 + t], false, false);
        }
#pragma unroll
        for (int t = 0; t < 4; ++t) bcur[t] = bnx[t];
        bp += (size_t)4 * Nc;
    }
    // peeled last k-step: no prefetch
    {
        v2f a0 = *(const v2f*)(afp0 + K - 4);
        v2f a1 = *(const v2f*)(afp1 + K - 4);
#pragma unroll
        for (int t = 0; t < 4; ++t) {
            acc[t]     = __builtin_amdgcn_wmma_f32_16x16x4_f32(
                             false, a0, false, bcur[t], (short)0, acc[t],     false, false);
            acc[4 + t] = __builtin_amdgcn_wmma_f32_16x16x4_f32(
                             false, a1, false, bcur[t], (short)0, acc[4 + t], false, false);
        }
    }

    // D layout: VGPR r -> M=base+r (lanes 0-15) / M=base+r+8 (lanes 16-31)
#pragma unroll
    for (int t = 0; t < 4; ++t) {
        int col = colw + t * 16 + lm;
        float bv = bias ? bias[col] : 0.0f;
#pragma unroll
        for (int r = 0; r < 8; ++r) {
            int row = row0 + r + half * 8;
            float v = acc[t][r] + bv;
            if (do_relu) v = fmaxf(v, 0.0f);
            if (resid) v += resid[(size_t)row * Nc + col];
            C[(size_t)row * Nc + col] = v;
        }
        if (two) {
#pragma unroll
            for (int r = 0; r < 8; ++r) {
                int row = row0 + 16 + r + half * 8;
                float v = acc[4 + t][r] + bv;
                if (do_relu) v = fmaxf(v, 0.0f);
                if (resid) v += resid[(size_t)row * Nc + col];
                C[(size_t)row * Nc + col] = v;
            }
        }
    }
}

// ---------- attention scores: as/ad[n,h] = dot(h[n,h,:], att_src/dst[h,:]) ----------
__global__ void att_scores(const float* __restrict__ h, const float* __restrict__ a_src,
                           const float* __restrict__ a_dst,
                           float* __restrict__ as_, float* __restrict__ ad_)
{
    int n = blockIdx.x;
    int head = threadIdx.y;
    int lane = threadIdx.x;
    const float* hp = h + (size_t)n * HC + head * CCH;
    const float* sp = a_src + head * CCH;
    const float* dp = a_dst + head * CCH;
    float ss = 0.f, sd = 0.f;
    for (int c = lane; c < CCH; c += 32) {
        float hv = hp[c];
        ss += hv * sp[c];
        sd += hv * dp[c];
    }
    for (int off = 16; off > 0; off >>= 1) {
        ss += __shfl_xor(ss, off);
        sd += __shfl_xor(sd, off);
    }
    if (lane == 0) {
        as_[n * HEADS + head] = ss;
        ad_[n * HEADS + head] = sd;
    }
}

// ---------- edge pass 1: leaky-relu logits + segment max ----------
__global__ void edge_logits_max(const int* __restrict__ src, const int* __restrict__ dst,
                                const float* __restrict__ as_, const float* __restrict__ ad_,
                                float* __restrict__ ebuf, unsigned* __restrict__ m_enc)
{
    int i = blockIdx.x * blockDim.x + threadIdx.x;
    if (i >= NEDGES * HEADS) return;
    int e = i >> 2, hd = i & 3;
    int s = src[e], d = dst[e];
    float v = as_[s * HEADS + hd] + ad_[d * HEADS + hd];
    v = (v > 0.f) ? v : 0.2f * v;                   // leaky_relu(0.2)
    ebuf[i] = v;
    atomicMax(&m_enc[d * HEADS + hd], enc_f32(v));
}

// ---------- edge pass 2: exp(e - m[dst]) + segment sum ----------
__global__ void edge_expsum(const int* __restrict__ dst, float* __restrict__ ebuf,
                            const unsigned* __restrict__ m_enc, float* __restrict__ denom)
{
    int i = blockIdx.x * blockDim.x + threadIdx.x;
    if (i >= NEDGES * HEADS) return;
    int e = i >> 2, hd = i & 3;
    int d = dst[e];
    float w = __expf(ebuf[i] - dec_f32(m_enc[d * HEADS + hd]));
    ebuf[i] = w;
    atomic_fadd(&denom[d * HEADS + hd], w);
}

// ---------- edge pass 3: agg[dst] += alpha * h[src] ----------
// wave covers 128 channels (one head) of one edge: coalesced float4 gather of h[src],
// coalesced hardware-fadd scatter into agg[dst] (L2-resident working set)
__global__ void edge_scatter(const int* __restrict__ src, const int* __restrict__ dst,
                             const float* __restrict__ ebuf, const float* __restrict__ denom,
                             const float* __restrict__ h, float* __restrict__ agg)
{
    int i = blockIdx.x * blockDim.x + threadIdx.x;
    if (i >= NEDGES * (HC / 4)) return;
    int e   = i >> 7;                               // HC/4 = 128 groups per edge
    int grp = i & 127;
    int c0  = grp * 4;
    int hd  = c0 >> 7;                              // channel block -> head (CCH=128)
    int s = src[e], d = dst[e];
    float alpha = ebuf[e * HEADS + hd] / (denom[d * HEADS + hd] + 1e-16f);
    float4 hv = *(const float4*)(h + (size_t)s * HC + c0);
    float* ap = agg + (size_t)d * HC + c0;
    atomic_fadd(ap + 0, alpha * hv.x);
    atomic_fadd(ap + 1, alpha * hv.y);
    atomic_fadd(ap + 2, alpha * hv.z);
    atomic_fadd(ap + 3, alpha * hv.w);
}

// ---------- y = relu(agg + gat_b), in place ----------
__global__ void bias_relu_inplace(float* __restrict__ agg, const float* __restrict__ b)
{
    int i = blockIdx.x * blockDim.x + threadIdx.x;
    if (i >= NNODES * HC) return;
    float v = agg[i] + b[i & (HC - 1)];
    agg[i] = fmaxf(v, 0.0f);
}

// ---------- global pooling ----------
__global__ void pool_accum(const float* __restrict__ x, const int* __restrict__ batch,
                           unsigned* __restrict__ gmax_enc, float* __restrict__ gsum)
{
    int i = blockIdx.x * blockDim.x + threadIdx.x;
    if (i >= NNODES * FIN) return;
    int n = i >> 7, c = i & 127;
    int g = batch[n];
    float v = x[i];
    atomicMax(&gmax_enc[g * FIN + c], enc_f32(v));
    atomic_fadd(&gsum[g * FIN + c], v);
}
__global__ void pool_count(const int* __restrict__ batch, float* __restrict__ cnt)
{
    int i = blockIdx.x * blockDim.x + threadIdx.x;
    if (i < NNODES) atomic_fadd(&cnt[batch[i]], 1.0f);
}
__global__ void pool_finalize(const unsigned* __restrict__ gmax_enc, const float* __restrict__ gsum,
                              const float* __restrict__ cnt, float* __restrict__ z)
{
    int i = blockIdx.x * blockDim.x + threadIdx.x;
    if (i >= GRAPHS * FIN) return;
    int g = i >> 7, c = i & 127;
    z[g * (2 * FIN) + c]       = dec_f32(gmax_enc[i]);
    z[g * (2 * FIN) + FIN + c] = gsum[i] / fmaxf(cnt[g], 1.0f);
}

// ---------- final [64,256] @ [256,10] + b3 (too narrow for WMMA) ----------
__global__ void final_dense(const float* __restrict__ z2, const float* __restrict__ w3,
                            const float* __restrict__ b3, float* __restrict__ out)
{
    int i = blockIdx.x * blockDim.x + threadIdx.x;
    if (i >= GRAPHS * NCLS) return;
    int g = i / NCLS, c = i % NCLS;
    float s = b3[c];
    for (int k = 0; k < DNN / 2; ++k)
        s += z2[g * (DNN / 2) + k] * w3[k * NCLS + c];
    out[i] = s;
}

static inline int cdiv(int a, int b) { return (a + b - 1) / b; }

extern "C" void kernel_launch(void* const* d_in, const int* in_sizes, int n_in,
                              void* d_out, int out_size, void* d_ws, size_t ws_size,
                              hipStream_t stream)
{
    const float* x_in    = (const float*)d_in[0];
    const float* gat_W   = (const float*)d_in[1];
    const float* att_src = (const float*)d_in[2];
    const float* att_dst = (const float*)d_in[3];
    const float* gat_b   = (const float*)d_in[4];
    const float* lin_W   = (const float*)d_in[5];
    const float* lin_b   = (const float*)d_in[6];
    const float* w1      = (const float*)d_in[7];
    const float* b1      = (const float*)d_in[8];
    const float* w2      = (const float*)d_in[9];
    const float* b2      = (const float*)d_in[10];
    const float* w3      = (const float*)d_in[11];
    const float* b3      = (const float*)d_in[12];
    const int*   eidx    = (const int*)d_in[13];
    const int*   batch   = (const int*)d_in[14];
    const int* srcI = eidx;            // edge_index[0,:]
    const int* dstI = eidx + NEDGES;   // edge_index[1,:]

    // ---- carve workspace ----
    char* ws = (char*)d_ws;
    size_t off = 0;
    auto carve = [&](size_t bytes) -> void* {
        void* p = ws + off;
        off = (off + bytes + 255) & ~(size_t)255;
        return p;
    };
    float*    h        = (float*)   carve((size_t)NNODES * HC * 4);
    float*    agg      = (float*)   carve((size_t)NNODES * HC * 4);
    float*    xb0      = (float*)   carve((size_t)NNODES * FIN * 4);
    float*    xb1      = (float*)   carve((size_t)NNODES * FIN * 4);
    float*    as_      = (float*)   carve((size_t)NNODES * HEADS * 4);
    float*    ad_      = (float*)   carve((size_t)NNODES * HEADS * 4);
    unsigned* m_enc    = (unsigned*)carve((size_t)NNODES * HEADS * 4);
    float*    denom    = (float*)   carve((size_t)NNODES * HEADS * 4);
    float*    ebuf     = (float*)   carve((size_t)NEDGES * HEADS * 4);
    unsigned* gmax_enc = (unsigned*)carve((size_t)GRAPHS * FIN * 4);
    float*    gsum     = (float*)   carve((size_t)GRAPHS * FIN * 4);
    float*    cnt      = (float*)   carve((size_t)GRAPHS * 4);
    float*    z        = (float*)   carve((size_t)GRAPHS * 2 * FIN * 4);
    float*    z1       = (float*)   carve((size_t)GRAPHS * DNN * 4);
    float*    z2       = (float*)   carve((size_t)GRAPHS * (DNN / 2) * 4);
    (void)ws_size; (void)in_sizes; (void)n_in; (void)out_size;

    const float* cur = x_in;
    float* nxt = xb0;
    for (int l = 0; l < LAYERS; ++l) {
        // h = cur @ gat_W[l]   [N,128] x [128,512]; 32-row panels
        wmma_gemm_f32<<<dim3(cdiv(NNODES, 32)), dim3(32, HC / 64), 0, stream>>>(
            cur, gat_W + (size_t)l * FIN * HC, nullptr, nullptr, h,
            NNODES, FIN, HC, 0);

        att_scores<<<NNODES, dim3(32, 4), 0, stream>>>(
            h, att_src + l * HEADS * CCH, att_dst + l * HEADS * CCH, as_, ad_);

        zero_u32<<<cdiv(NNODES * HEADS, 256), 256, 0, stream>>>(m_enc, NNODES * HEADS);
        zero_u32<<<cdiv(NNODES * HEADS, 256), 256, 0, stream>>>((unsigned*)denom, NNODES * HEADS);
        zero_u32<<<cdiv(NNODES * HC, 256), 256, 0, stream>>>((unsigned*)agg, NNODES * HC);

        edge_logits_max<<<cdiv(NEDGES * HEADS, 256), 256, 0, stream>>>(
            srcI, dstI, as_, ad_, ebuf, m_enc);
        edge_expsum<<<cdiv(NEDGES * HEADS, 256), 256, 0, stream>>>(
            dstI, ebuf, m_enc, denom);
        edge_scatter<<<cdiv(NEDGES * (HC / 4), 256), 256, 0, stream>>>(
            srcI, dstI, ebuf, denom, h, agg);

        bias_relu_inplace<<<cdiv(NNODES * HC, 256), 256, 0, stream>>>(agg, gat_b + l * HC);

        // nxt = relu(agg @ lin_W[l] + lin_b[l]) [+ cur if l>0]
        const float* res = (l > 0) ? cur : nullptr;
        wmma_gemm_f32<<<dim3(cdiv(NNODES, 32)), dim3(32, FIN / 64), 0, stream>>>(
            agg, lin_W + (size_t)l * HC * FIN, lin_b + l * FIN, res, nxt,
            NNODES, HC, FIN, 1);

        cur = nxt;
        nxt = (nxt == xb0) ? xb1 : xb0;
    }

    // ---- pooling: max + mean per graph, concat -> z[G, 256] ----
    zero_u32<<<cdiv(GRAPHS * FIN, 256), 256, 0, stream>>>(gmax_enc, GRAPHS * FIN);
    zero_u32<<<cdiv(GRAPHS * FIN, 256), 256, 0, stream>>>((unsigned*)gsum, GRAPHS * FIN);
    zero_u32<<<1, 256, 0, stream>>>((unsigned*)cnt, GRAPHS);
    pool_accum<<<cdiv(NNODES * FIN, 256), 256, 0, stream>>>(cur, batch, gmax_enc, gsum);
    pool_count<<<cdiv(NNODES, 256), 256, 0, stream>>>(batch, cnt);
    pool_finalize<<<cdiv(GRAPHS * FIN, 256), 256, 0, stream>>>(gmax_enc, gsum, cnt, z);

    // ---- MLP head ----
    wmma_gemm_f32<<<dim3(GRAPHS / 32), dim3(32, DNN / 64), 0, stream>>>(
        z, w1, b1, nullptr, z1, GRAPHS, 2 * FIN, DNN, 1);
    wmma_gemm_f32<<<dim3(GRAPHS / 32), dim3(32, (DNN / 2) / 64), 0, stream>>>(
        z1, w2, b2, nullptr, z2, GRAPHS, DNN, DNN / 2, 1);
    final_dense<<<cdiv(GRAPHS * NCLS, 256), 256, 0, stream>>>(z2, w3, b3, (float*)d_out);
}